// Dwt_25331717111850
// MI455X (gfx1250) — compile-verified
//
#include <hip/hip_runtime.h>

// Haar 2x2 block DWT, (16,3,1024,1024) f32 -> (16,12,512,512) f32.
// Bandwidth-bound (402 MB stream, ~17us floor @ 23.3 TB/s). Transform computed
// on the matrix pipe via V_WMMA_F32_16X16X4_F32 (K=4 matches the 4-point block
// transform exactly, full fp32). Data is loaded directly in WMMA B-layout with
// a single coalesced b64 load per wave per tile -- zero cross-lane shuffles.

typedef float v2f __attribute__((ext_vector_type(2)));
typedef float v8f __attribute__((ext_vector_type(8)));

#ifndef __has_builtin
#define __has_builtin(x) 0
#endif

namespace {
constexpr int BATCH = 16, CH = 3, H = 1024, W = 1024;
constexpr int HO = H / 2, WO = W / 2;        // 512 x 512 per band
constexpr int TILES = WO / 16;               // 32 tiles of 16 blocks per row-pair
constexpr int TOTAL_WAVES = BATCH * CH * HO; // 24576 (one wave per output row)
constexpr int WAVES_PER_BLOCK = 8;           // 256 threads (8 x wave32)
}

__global__ __launch_bounds__(256) void haar_dwt_wmma(const float* __restrict__ xin,
                                                     float* __restrict__ yout) {
  const int lane = threadIdx.x & 31;
  const int l15  = lane & 15;          // B/D column within tile
  const int half = lane >> 4;          // 0: K=0,1 rows / bands LL,LH ; 1: K=2,3 / HL,HH
  const int m    = lane & 3;           // Haar band index for A rows (rows = 4r+m)

  const int wave = blockIdx.x * WAVES_PER_BLOCK + (threadIdx.x >> 5);
  const int i  = wave % HO;            // output row
  const int bc = wave / HO;            // fused (batch, channel)
  const int c  = bc % CH;
  const int b  = bc / CH;

  // ---- A matrix (16x4 f32). Band rows (on block values a,b,c,d):
  //   LL' = 0.25*(a+b+c+d)   (compensate 0.5 folded in)
  //   LH  = 0.5*(a+b-c-d); HL = 0.5*(a-b+c-d); HH = 0.5*(a-b-c+d)
  // ISA layout: lanes0-15 -> {A[L,0],A[L,1]}, lanes16-31 -> {A[L-16,2],A[L-16,3]}
  const int c0 = half ? 2 : 0;         // which K column this half-wave supplies (even slot)
  const int c1 = half ? 3 : 1;         // (odd slot)
  v2f av;
  av.x = (m == 0) ? 0.25f : ((c0 == 0 || c0 == m) ? 0.5f : -0.5f);
  av.y = (m == 0) ? 0.25f : ((c1 == m) ? 0.5f : -0.5f);

  // ---- Input: lane n reads float2 (a,b) of block n from row 2i;
  //             lane 16+n reads float2 (c,d) of block n from row 2i+1.
  // => one global_load_b64 per tile, two dense 128B runs, exactly WMMA B-layout.
  const float* pin = xin + ((size_t)bc * H + (size_t)(2 * i + half)) * (size_t)W
                         + (size_t)(2 * l15);

  // ---- Output planes: low plane = c ; high planes = 3+3c (LH), 4+3c (HL), 5+3c (HH)
  const int pA = half ? (4 + 3 * c) : c;            // HL : LL
  const int pB = half ? (5 + 3 * c) : (3 + 3 * c);  // HH : LH
  float* poutA = yout + (((size_t)b * 12 + pA) * HO + i) * (size_t)WO + l15;
  float* poutB = yout + (((size_t)b * 12 + pB) * HO + i) * (size_t)WO + l15;

#pragma unroll 8
  for (int j = 0; j < TILES; ++j) {
    // Non-temporal streaming load (402MB stream; keep L2 clean).
    const v2f bv = __builtin_nontemporal_load((const v2f*)(pin + 32 * j));

    float t0, t1;
#if __has_builtin(__builtin_amdgcn_wmma_f32_16x16x4_f32)
    // D(16x16,f32) = A(16x4) x B(4x16) + 0. A rows repeat every 4 -> every lane's
    // d[0..3] are the 4 bands of its block (rows v resp. v+8 of D).
    v8f acc = {0.f, 0.f, 0.f, 0.f, 0.f, 0.f, 0.f, 0.f};
    v8f d = __builtin_amdgcn_wmma_f32_16x16x4_f32(
        /*neg_a=*/false, av, /*neg_b=*/false, bv,
        /*c_mod=*/(short)0, acc, /*reuse_a=*/false, /*reuse_b=*/false);
    t0 = half ? d[2] : d[0];   // lanes0-15: LL'   lanes16-31: HL
    t1 = half ? d[3] : d[1];   // lanes0-15: LH    lanes16-31: HH
#else
    // VALU fallback: exchange halves to gather the full 2x2 block per lane.
    float px = __shfl_xor(bv.x, 16, 32);
    float py = __shfl_xor(bv.y, 16, 32);
    float a  = half ? px : bv.x, bb = half ? py : bv.y;
    float cc = half ? bv.x : px, dd = half ? bv.y : py;
    float band0 = 0.25f * (a + bb + cc + dd);
    float band1 = 0.5f  * (a + bb - cc - dd);
    float band2 = 0.5f  * (a - bb + cc - dd);
    float band3 = 0.5f  * (a - bb - cc + dd);
    t0 = half ? band2 : band0;
    t1 = half ? band3 : band1;
#endif
    __builtin_nontemporal_store(t0, poutA + 16 * j);
    __builtin_nontemporal_store(t1, poutB + 16 * j);
  }
}

extern "C" void kernel_launch(void* const* d_in, const int* in_sizes, int n_in,
                              void* d_out, int out_size, void* d_ws, size_t ws_size,
                              hipStream_t stream) {
  (void)in_sizes; (void)n_in; (void)d_ws; (void)ws_size; (void)out_size;
  const float* x = (const float*)d_in[0];
  float* out = (float*)d_out;
  dim3 grid(TOTAL_WAVES / WAVES_PER_BLOCK);  // 3072 blocks
  dim3 block(WAVES_PER_BLOCK * 32);          // 256 threads = 8 waves
  haar_dwt_wmma<<<grid, block, 0, stream>>>(x, out);
}